// ReconstructionLoss_14766097563655
// MI455X (gfx1250) — compile-verified
//
#include <hip/hip_runtime.h>
#include <stdint.h>

// ---------------- SSIM + L1 reconstruction loss, fused single pass ----------
// Shapes fixed by the reference: B=32, C=3, H=384, W=640, fp32.
// Memory-bound (~220 MB @ 23.3 TB/s ~ 9.5 us floor). No matmul structure ->
// WMMA not applicable; the CDNA5 win is the async global->LDS DMA path
// (ASYNCcnt, in-order retirement) double-buffered across the channel loop,
// plus b128 non-temporal output stores.

#define TILE_W   64
#define TILE_H   16
#define PH       (TILE_H + 2)      // 18 padded rows
#define PC       (TILE_W + 2)      // 66 padded cols (logical)
#define PWL      72                // LDS row stride (floats): interior 16B-aligned
#define COL_OFF  3                 // logical padded col p stored at index p+3
#define NTHREADS 256
#define N_HALO   164               // 2*66 (top/bottom rows) + 16*2 (side cols)

#define C1_CONST 6.5025f           // (0.01*255)^2
#define C2_CONST 58.5225f          // (0.03*255)^2
#define W_SSIM   0.425f            // 0.5 * ALPHA
#define W_L1     0.15f             // 1 - ALPHA

typedef float v4f __attribute__((ext_vector_type(4)));

__device__ __forceinline__ int reflect_idx(int i, int n) {
    i = (i < 0) ? -i : i;                // pad=1: -1 -> 1
    i = (i >= n) ? (2 * n - 2 - i) : i;  // pad=1:  n -> n-2
    return i;
}

// Async DMA: global -> LDS, per-lane, tracked on ASYNCcnt (in-order retire).
__device__ __forceinline__ void async_ld_b32(uint32_t lds_addr, uint64_t gaddr) {
    asm volatile("global_load_async_to_lds_b32 %0, %1, off"
                 :: "v"(lds_addr), "v"(gaddr) : "memory");
}
__device__ __forceinline__ void async_ld_b128(uint32_t lds_addr, uint64_t gaddr) {
    asm volatile("global_load_async_to_lds_b128 %0, %1, off"
                 :: "v"(lds_addr), "v"(gaddr) : "memory");
}
template <int N>
__device__ __forceinline__ void wait_async() {
    asm volatile("s_wait_asynccnt %0" :: "i"(N) : "memory");
}

// Issue all async loads for one channel into one LDS buffer.
// Every wave issues exactly 4 async ops (uniform ASYNCcnt bookkeeping).
__device__ __forceinline__ void issue_channel(
    const float* __restrict__ sBase, const float* __restrict__ tBase,
    uint32_t ldsS, uint32_t ldsT, int tid, int ty0, int tx0, int H, int W)
{
    // ---- interior: 64x16, one B128 per thread per tensor, 16B aligned ----
    {
        const int ir = tid >> 4;          // 0..15 tile row
        const int iq = tid & 15;          // 0..15 quad within row
        const int pr = 1 + ir;
        const int pc = 1 + iq * 4;
        const uint32_t la = (uint32_t)((pr * PWL + pc + COL_OFF) * 4);
        const size_t gi = (size_t)(ty0 + ir) * W + (size_t)(tx0 + iq * 4);
        async_ld_b128(ldsS + la, (uint64_t)(uintptr_t)(sBase + gi));
        async_ld_b128(ldsT + la, (uint64_t)(uintptr_t)(tBase + gi));
    }
    // ---- halo: 164 reflected elems; threads >=164 redo elems 0..91 (benign
    //      duplicate writes of identical values) so every wave issues 4 ops ----
    {
        int i = tid - ((tid >= N_HALO) ? N_HALO : 0);
        int pr, pc;
        if (i < 2 * PC) {                 // top & bottom padded rows
            pr = (i < PC) ? 0 : (PH - 1);
            pc = (i < PC) ? i : (i - PC);
        } else {                          // side columns, rows 1..16
            const int j = i - 2 * PC;
            pr = 1 + (j >> 1);
            pc = (j & 1) ? (PC - 1) : 0;
        }
        const int gr = reflect_idx(ty0 + pr - 1, H);
        const int gc = reflect_idx(tx0 + pc - 1, W);
        const uint32_t la = (uint32_t)((pr * PWL + pc + COL_OFF) * 4);
        const size_t gi = (size_t)gr * W + (size_t)gc;
        async_ld_b32(ldsS + la, (uint64_t)(uintptr_t)(sBase + gi));
        async_ld_b32(ldsT + la, (uint64_t)(uintptr_t)(tBase + gi));
    }
}

// Compute 4 adjacent output pixels (one row, cols col4..col4+3) for one channel.
__device__ __forceinline__ v4f compute_channel(
    const float* __restrict__ sS, const float* __restrict__ sT, int row, int col4)
{
    float Sx[4]  = {0.f, 0.f, 0.f, 0.f};
    float Sy[4]  = {0.f, 0.f, 0.f, 0.f};
    float Sxx[4] = {0.f, 0.f, 0.f, 0.f};
    float Syy[4] = {0.f, 0.f, 0.f, 0.f};
    float Sxy[4] = {0.f, 0.f, 0.f, 0.f};
    float cs[4], ct[4];

    #pragma unroll
    for (int r = 0; r < 3; ++r) {
        const int base = (row + r) * PWL + col4 + COL_OFF;
        float s[6], t[6];
        #pragma unroll
        for (int k = 0; k < 6; ++k) { s[k] = sS[base + k]; t[k] = sT[base + k]; }
        if (r == 1) {
            #pragma unroll
            for (int j = 0; j < 4; ++j) { cs[j] = s[j + 1]; ct[j] = t[j + 1]; }
        }
        #pragma unroll
        for (int j = 0; j < 4; ++j) {
            Sx[j]  += s[j] + s[j + 1] + s[j + 2];
            Sy[j]  += t[j] + t[j + 1] + t[j + 2];
            Sxx[j] += s[j] * s[j] + s[j + 1] * s[j + 1] + s[j + 2] * s[j + 2];
            Syy[j] += t[j] * t[j] + t[j + 1] * t[j + 1] + t[j + 2] * t[j + 2];
            Sxy[j] += s[j] * t[j] + s[j + 1] * t[j + 1] + s[j + 2] * t[j + 2];
        }
    }

    v4f res;
    #pragma unroll
    for (int j = 0; j < 4; ++j) {
        const float inv9 = (1.0f / 9.0f);
        const float mu_x = Sx[j]  * inv9;
        const float mu_y = Sy[j]  * inv9;
        const float Exx  = Sxx[j] * inv9;
        const float Eyy  = Syy[j] * inv9;
        const float Exy  = Sxy[j] * inv9;
        const float sig_x  = Exx - mu_x;          // faithful: minus mu, not mu^2
        const float sig_y  = Eyy - mu_y;
        const float sig_xy = Exy - mu_x * mu_y;
        const float num = (2.f * mu_x * mu_y + C1_CONST) * (2.f * sig_xy + C2_CONST);
        const float den = (mu_x * mu_x + mu_y * mu_y + C1_CONST) * (sig_x + sig_y + C2_CONST);
        float ssim = 0.5f * (1.f - num / den);
        ssim = fminf(fmaxf(ssim, 0.f), 1.f);
        const float l1 = fabsf(cs[j] - ct[j]);
        res[j] = W_SSIM * ssim + W_L1 * l1;
    }
    return res;
}

__global__ __launch_bounds__(NTHREADS) void recon_loss_kernel(
    const float* __restrict__ src, const float* __restrict__ tgt,
    float* __restrict__ out, int H, int W)
{
    __shared__ float sS[2][PH * PWL];
    __shared__ float sT[2][PH * PWL];

    const int tx0 = blockIdx.x * TILE_W;
    const int ty0 = blockIdx.y * TILE_H;
    const int b   = blockIdx.z;
    const int tid = threadIdx.x;

    const int row  = tid >> 4;            // 0..15 output row in tile
    const int col4 = (tid & 15) * 4;      // 0,4,...,60 first output col

    const uint32_t ldsS0 = (uint32_t)(uintptr_t)&sS[0][0];
    const uint32_t ldsS1 = (uint32_t)(uintptr_t)&sS[1][0];
    const uint32_t ldsT0 = (uint32_t)(uintptr_t)&sT[0][0];
    const uint32_t ldsT1 = (uint32_t)(uintptr_t)&sT[1][0];

    const size_t planeSz = (size_t)H * (size_t)W;
    const float* s0 = src + (size_t)(b * 3 + 0) * planeSz;
    const float* s1 = src + (size_t)(b * 3 + 1) * planeSz;
    const float* s2 = src + (size_t)(b * 3 + 2) * planeSz;
    const float* t0 = tgt + (size_t)(b * 3 + 0) * planeSz;
    const float* t1 = tgt + (size_t)(b * 3 + 1) * planeSz;
    const float* t2 = tgt + (size_t)(b * 3 + 2) * planeSz;

    // ---- software pipeline: prefetch c0 -> buf0, c1 -> buf1 ----
    issue_channel(s0, t0, ldsS0, ldsT0, tid, ty0, tx0, H, W);   // 4 async ops
    issue_channel(s1, t1, ldsS1, ldsT1, tid, ty0, tx0, H, W);   // 4 async ops

    // ---- channel 0 ----
    wait_async<4>();                      // oldest 4 (c0) retired, c1 in flight
    __syncthreads();
    v4f acc = compute_channel(sS[0], sT[0], row, col4);
    __syncthreads();                      // buf0 free: safe to overwrite

    // ---- prefetch c2 -> buf0, compute channel 1 ----
    issue_channel(s2, t2, ldsS0, ldsT0, tid, ty0, tx0, H, W);   // 4 async ops
    wait_async<4>();                      // c1 retired (in-order), c2 in flight
    __syncthreads();
    {
        v4f r = compute_channel(sS[1], sT[1], row, col4);
        acc += r;
    }

    // ---- channel 2 ----
    wait_async<0>();                      // c2 retired
    __syncthreads();
    {
        v4f r = compute_channel(sS[0], sT[0], row, col4);
        acc += r;
    }

    // ---- channel mean + one b128 non-temporal store per thread ----
    acc *= (1.0f / 3.0f);
    float* o = out + ((size_t)b * H + (size_t)(ty0 + row)) * W + (size_t)(tx0 + col4);
    __builtin_nontemporal_store(acc, (v4f*)o);
}

extern "C" void kernel_launch(void* const* d_in, const int* in_sizes, int n_in,
                              void* d_out, int out_size, void* d_ws, size_t ws_size,
                              hipStream_t stream) {
    (void)in_sizes; (void)n_in; (void)out_size; (void)d_ws; (void)ws_size;
    const float* src = (const float*)d_in[0];   // 'output' in reference
    const float* tgt = (const float*)d_in[1];   // 'target'
    float* out = (float*)d_out;                 // [32,1,384,640] fp32

    const int B = 32, H = 384, W = 640;
    dim3 grid(W / TILE_W, H / TILE_H, B);       // 10 x 24 x 32
    recon_loss_kernel<<<grid, NTHREADS, 0, stream>>>(src, tgt, out, H, W);
}